// NeuronPool_50680614092898
// MI455X (gfx1250) — compile-verified
//
#include <hip/hip_runtime.h>
#include <math.h>

// Problem constants (from reference): POOL=512, DM=256, DF=64, B*S*K=8192
#define POOLC 512
#define DMC   256
#define DFC   64

typedef __attribute__((ext_vector_type(2))) float v2f;
typedef __attribute__((ext_vector_type(8))) float v8f;
typedef __attribute__((ext_vector_type(4))) unsigned int v4u;
typedef __attribute__((ext_vector_type(8))) int v8i;
typedef __attribute__((ext_vector_type(4))) int v4i;

// ---------------- Phase 1: bucket assignments by expert ----------------

__global__ void np_zero(int* __restrict__ p, int n) {
  int i = blockIdx.x * blockDim.x + threadIdx.x;
  if (i < n) p[i] = 0;
}

__global__ void np_count(const int* __restrict__ idx, int* __restrict__ cnt, int na) {
  int i = blockIdx.x * blockDim.x + threadIdx.x;
  if (i < na) atomicAdd(&cnt[idx[i] & (POOLC - 1)], 1);
}

// Exclusive scan over POOLC counters, single 512-thread block (Hillis-Steele).
__global__ void np_scan(const int* __restrict__ cnt, int* __restrict__ off) {
  __shared__ int tmp[POOLC];
  int t = threadIdx.x;
  tmp[t] = cnt[t];
  __syncthreads();
  for (int s = 1; s < POOLC; s <<= 1) {
    int v = (t >= s) ? tmp[t - s] : 0;
    __syncthreads();
    tmp[t] += v;
    __syncthreads();
  }
  off[t + 1] = tmp[t];
  if (t == 0) off[0] = 0;
}

__global__ void np_scatter(const int* __restrict__ idx, const int* __restrict__ off,
                           int* __restrict__ cur, int* __restrict__ bucket, int na) {
  int i = blockIdx.x * blockDim.x + threadIdx.x;
  if (i < na) {
    int e = idx[i] & (POOLC - 1);
    int p = atomicAdd(&cur[e], 1);
    bucket[off[e] + p] = i;
  }
}

// ---------------- TDM descriptor helpers (CDNA5 ISA) ----
// D# group0: [1:0]=count, [30]=gather_index_size, [31]=gather_mode,
//            [63:32]=lds_addr, [120:64]=global_addr, [127:126]=type(2)
// D# group1: [17:16]=data_size, [20]=pad_enable, [24:22]=pad_interval,
//            [31:25]=pad_amount, [79:48]=tensor_dim0, [111:80]=tensor_dim1,
//            [127:112]=tile_dim0, [143:128]=tile_dim1, [207:160]=dim0_stride

// data_size=4B | pad_enable | pad every 64 dwords | pad 8 dwords -> stride 72
#define W0_WIN ((2u << 16) | (1u << 20) | (5u << 22) | (7u << 25))
// data_size=4B | pad_enable | pad every 256 dwords | pad 4 dwords -> stride 260
#define W0_XS  ((2u << 16) | (1u << 20) | (7u << 22) | (3u << 25))

__device__ __forceinline__ void tdm_issue(v4u g0, v8i g1, v4i g2, v4i g3) {
  // This toolchain exposes the 6-arg builtin:
  // (uint32x4 g0, int32x8 g1, int32x4 g2, int32x4 g3, int32x8, i32 cpol)
  v8i z8 = {0, 0, 0, 0, 0, 0, 0, 0};
  __builtin_amdgcn_tensor_load_to_lds(g0, g1, g2, g3, z8, 0);
}

__device__ __forceinline__ void tdm_load_2d(unsigned lds_off, const void* gaddr,
                                            unsigned w0, unsigned td0, unsigned td1,
                                            unsigned tile0, unsigned tile1,
                                            unsigned stride0) {
  unsigned long long ga = (unsigned long long)(uintptr_t)gaddr;
  v4u g0;
  g0.x = 1u;                                   // count=1, user descriptor
  g0.y = lds_off;
  g0.z = (unsigned)ga;
  g0.w = (unsigned)(ga >> 32) | (2u << 30);    // type=2 ("image")
  v8i g1;
  g1[0] = (int)w0;
  g1[1] = (int)(td0 << 16);
  g1[2] = (int)((td0 >> 16) | (td1 << 16));
  g1[3] = (int)((td1 >> 16) | (tile0 << 16));
  g1[4] = (int)(tile1 & 0xFFFFu);
  g1[5] = (int)stride0;
  g1[6] = 0;
  g1[7] = 0;
  v4i z = {0, 0, 0, 0};
  tdm_issue(g0, g1, z, z);
}

// Gather mode: one TDM fetches 16 scattered rows (16-bit row indices in g2/g3).
__device__ __forceinline__ void tdm_gather16(unsigned lds_off, const void* gbase,
                                             unsigned rowlen, unsigned nrows,
                                             const int* rowidx) {
  unsigned long long ga = (unsigned long long)(uintptr_t)gbase;
  v4u g0;
  g0.x = 1u | (1u << 31);                      // count=1, gather_mode, 16-bit idx
  g0.y = lds_off;
  g0.z = (unsigned)ga;
  g0.w = (unsigned)(ga >> 32) | (2u << 30);
  v8i g1;
  g1[0] = (int)W0_XS;
  g1[1] = (int)(rowlen << 16);
  g1[2] = (int)((rowlen >> 16) | (nrows << 16));
  g1[3] = (int)((nrows >> 16) | (rowlen << 16));  // tile_dim0 = rowlen
  g1[4] = 16;                                     // tile_dim1 = #indices
  g1[5] = (int)rowlen;                            // dim0 stride
  g1[6] = 0;
  g1[7] = 0;
  v4i g2, g3;
#pragma unroll
  for (int i = 0; i < 4; ++i)
    g2[i] = (rowidx[2 * i] & 0xFFFF) | ((rowidx[2 * i + 1] & 0xFFFF) << 16);
#pragma unroll
  for (int i = 0; i < 4; ++i)
    g3[i] = (rowidx[8 + 2 * i] & 0xFFFF) | ((rowidx[9 + 2 * i] & 0xFFFF) << 16);
  tdm_issue(g0, g1, g2, g3);
}

// ---------------- Phase 2: per-expert WMMA GEMM ----------------
// One workgroup per expert, 4 waves (128 threads, wave32).
// Chunks of 16 tokens: hidden[16,64] = xs[16,256] @ w_in[e];  gelu;
//                      out  [16,256] = hs[16,64]  @ w_out[e]
// WMMA f32 16x16x4 fragments:
//   A: lane L<16 -> row M=L, K={0,1}; lane L+16 -> row M=L, K={2,3}
//   B: lane L    -> col N=L%16, same K split
//   C/D: vgpr i  -> row M=i (lanes 0-15) / M=i+8 (lanes 16-31), col N=lane%16
#define XS_STRIDE 260   // TDM-padded; 260 % 64 == 4 -> conflict-free A reads
#define HS_STRIDE 68    // 68 % 64 == 4
#define WS_STRIDE 72    // TDM-padded; 72 % 32 == 8 -> half-waves hit disjoint banks
#define KCHUNK    64    // w_in rows staged per TDM

__global__ __launch_bounds__(128) void np_expert_gemm(
    const float* __restrict__ x, const float* __restrict__ w_in,
    const float* __restrict__ w_out, const float* __restrict__ b_in,
    const float* __restrict__ b_out, const int* __restrict__ bucket,
    const int* __restrict__ off, float* __restrict__ out, int kway, int ntok)
{
  const int e   = blockIdx.x;
  const int beg = off[e];
  const int end = off[e + 1];
  if (beg == end) return;

  const int tid   = threadIdx.x;
  const int wave  = tid >> 5;         // 0..3
  const int lane  = tid & 31;
  const int ln    = lane & 15;
  const int khalf = (lane >> 4) << 1; // 0 or 2

  __shared__ __attribute__((aligned(16))) float xs[16 * XS_STRIDE];
  __shared__ __attribute__((aligned(16))) float hs[16 * HS_STRIDE];
  __shared__ __attribute__((aligned(16))) float wbuf[2][KCHUNK * WS_STRIDE];
  __shared__ int aidx[16];   // assignment id (output row), -1 = padded
  __shared__ int atok[16];   // token id (x row), clamped to 0 for padded

  const float* wi = w_in  + (size_t)e * DMC * DFC;
  const float* wo = w_out + (size_t)e * DFC * DMC;

  const unsigned xs_off = (unsigned)(uintptr_t)&xs[0];
  const unsigned wb_off[2] = { (unsigned)(uintptr_t)&wbuf[0][0],
                               (unsigned)(uintptr_t)&wbuf[1][0] };

  for (int base = beg; base < end; base += 16) {
    if (tid < 16) {
      int p = base + tid;
      int a = (p < end) ? bucket[p] : -1;
      aidx[tid] = a;
      atok[tid] = (a >= 0) ? (a / kway) : 0;
    }
    __syncthreads();

    // Wave 0 launches the DMA engine: gather 16 scattered x rows with one
    // gather-mode TDM, plus the first 64-row w_in chunk (double-buffered below).
    if (wave == 0) {
      tdm_gather16(xs_off, x, DMC, (unsigned)ntok, atok);
      tdm_load_2d(wb_off[0], wi, W0_WIN, DFC, KCHUNK, DFC, KCHUNK, DFC);
    }
    // Warm the L2->L0 path for the w_out tile needed by GEMM2.
    for (int p = tid; p < (DFC * DMC) / 32; p += 128)
      __builtin_prefetch(wo + p * 32, 0, 0);

    // ---- GEMM1: hidden tile; wave w owns cols [w*16, w*16+16) ----
    v8f acc = {};
    const int col = wave * 16 + ln;
    for (int kc = 0; kc < DMC / KCHUNK; ++kc) {
      if (wave == 0) __builtin_amdgcn_s_wait_tensorcnt(0);
      __syncthreads();   // xs + wbuf[kc&1] visible to all waves
      if (wave == 0 && kc + 1 < DMC / KCHUNK)
        tdm_load_2d(wb_off[(kc + 1) & 1], wi + (size_t)(kc + 1) * KCHUNK * DFC,
                    W0_WIN, DFC, KCHUNK, DFC, KCHUNK, DFC);
      const float* wl = &wbuf[kc & 1][0];
#pragma unroll
      for (int kk = 0; kk < KCHUNK; kk += 4) {
        const int kr = kk + khalf;
        v2f a = *(const v2f*)(&xs[ln * XS_STRIDE + kc * KCHUNK + kr]);
        v2f b;
        b.x = wl[kr * WS_STRIDE + col];
        b.y = wl[(kr + 1) * WS_STRIDE + col];
        acc = __builtin_amdgcn_wmma_f32_16x16x4_f32(
            false, a, false, b, (short)0, acc, false, false);
      }
      __syncthreads();   // all reads of wbuf[kc&1] done before its next DMA
    }

    // bias + exact-erf GELU, spill hidden tile to LDS
    {
      float bi = b_in[e * DFC + wave * 16 + ln];
      int rbase = (lane < 16) ? 0 : 8;
#pragma unroll
      for (int i = 0; i < 8; ++i) {
        float h = acc[i] + bi;
        h = 0.5f * h * (1.0f + erff(h * 0.70710678118654752f));
        hs[(rbase + i) * HS_STRIDE + wave * 16 + ln] = h;
      }
    }
    __syncthreads();

    // ---- GEMM2: out tile; wave w owns N-tiles w*4 .. w*4+3 ----
    for (int j = 0; j < 4; ++j) {
      const int nt = wave * 4 + j;
      v8f acc2 = {};
      const float* wb = wo + nt * 16 + ln;
#pragma unroll
      for (int kk = 0; kk < DFC; kk += 4) {
        v2f a = *(const v2f*)(&hs[ln * HS_STRIDE + kk + khalf]);
        v2f b;
        b.x = wb[(size_t)(kk + khalf)     * DMC];
        b.y = wb[(size_t)(kk + khalf + 1) * DMC];
        acc2 = __builtin_amdgcn_wmma_f32_16x16x4_f32(
            false, a, false, b, (short)0, acc2, false, false);
      }
      float bo = b_out[e * DMC + nt * 16 + ln];
      int rbase = (lane < 16) ? 0 : 8;
#pragma unroll
      for (int i = 0; i < 8; ++i) {
        int a = aidx[rbase + i];
        if (a >= 0) out[(size_t)a * DMC + nt * 16 + ln] = acc2[i] + bo;
      }
    }
    __syncthreads();
  }
}

// ---------------- Launch ----------------

extern "C" void kernel_launch(void* const* d_in, const int* in_sizes, int n_in,
                              void* d_out, int out_size, void* d_ws, size_t ws_size,
                              hipStream_t stream) {
  const float* x     = (const float*)d_in[0];
  const int*   idx   = (const int*)  d_in[1];
  const float* w_in  = (const float*)d_in[2];
  const float* w_out = (const float*)d_in[3];
  const float* b_in  = (const float*)d_in[4];
  const float* b_out = (const float*)d_in[5];
  float* out = (float*)d_out;

  const int na     = in_sizes[1];           // B*S*K assignments
  const int tokens = in_sizes[0] / DMC;     // B*S
  const int kway   = na / tokens;           // K

  int* cnt    = (int*)d_ws;                 // [POOLC]
  int* cur    = cnt + POOLC;                // [POOLC]
  int* off    = cur + POOLC;                // [POOLC+1]
  int* bucket = off + POOLC + 1;            // [na]

  np_zero   <<<(2 * POOLC + 255) / 256, 256, 0, stream>>>(cnt, 2 * POOLC);
  np_count  <<<(na + 255) / 256,        256, 0, stream>>>(idx, cnt, na);
  np_scan   <<<1, POOLC, 0, stream>>>(cnt, off);
  np_scatter<<<(na + 255) / 256,        256, 0, stream>>>(idx, off, cur, bucket, na);
  np_expert_gemm<<<POOLC, 128, 0, stream>>>(x, w_in, w_out, b_in, b_out,
                                            bucket, off, out, kway, tokens);
}